// LangVisNet_46531675685247
// MI455X (gfx1250) — compile-verified
//
#include <hip/hip_runtime.h>
#include <hip/hip_bf16.h>
#include <math.h>

// ---------------- types ----------------
typedef float v8f __attribute__((ext_vector_type(8)));
typedef float v2f __attribute__((ext_vector_type(2)));

// ---------------- constants ----------------
#define EMB   1000
#define HID   1000
#define VIS   2688
#define C2    2690          // VIS + 2 (x,y grids)
#define MIX   2688
#define HMIX  1005
#define G4MIX 4020          // 4*HMIX
#define CCLD  4691          // cc_W row length
#define TT    8
#define HH    16
#define WW    16
#define HW    256
#define SEQ   2048          // HW*T
#define HSTR  1008          // padded hseq row stride (8B-aligned rows)
#define RSTR  1006          // padded Whh slab row stride in LDS
#define SCAN_BLOCKS   64
#define SCAN_THREADS  256

__device__ __forceinline__ float sigm(float x) { return 1.0f / (1.0f + __expf(-x)); }

// ================= embedding gather =================
__global__ void k_embed(const int* __restrict__ lang, const float* __restrict__ emb,
                        float* __restrict__ le) {
    int idx = blockIdx.x * blockDim.x + threadIdx.x;
    if (idx >= TT * EMB) return;
    int t = idx / EMB, e = idx % EMB;
    le[idx] = emb[(long)lang[t] * EMB + e];
}

// ================= generic fp32 WMMA GEMM (16x64 tile / wave) =================
// C[m,n] = sum_k A[m*lda+k] * B[k*ldbk + n*ldbn]  (+biasN +biasN2 +biasM) (+=C if beta)
// AM: 0 = scalar-pair A loads (4B-aligned rows), 1 = v2f (8B-aligned rows)
// BM: 0 = strided (runtime ldbk), 1 = ldbk==1 scalar-pair, 2 = ldbk==1 v2f (aligned)
template<int AM, int BM>
__global__ void k_wmma_gemm_t(const float* __restrict__ A, int lda,
                              const float* __restrict__ B, long ldbk, long ldbn,
                              float* __restrict__ C, int ldc,
                              const float* __restrict__ biasN,
                              const float* __restrict__ biasN2,
                              const float* __restrict__ biasM,
                              int M, int N, int K, int beta) {
    int mt = (M + 15) >> 4;
    int nt = (N + 63) >> 6;
    int wave = blockIdx.x * (blockDim.x >> 5) + (threadIdx.x >> 5);
    if (wave >= mt * nt) return;               // wave-uniform: EXEC stays all-ones
    int tm = (wave / nt) << 4;
    int tn = (wave % nt) << 6;

    int lane = threadIdx.x & 31;
    int lr = lane & 15;        // row/col within half
    int lh = lane >> 4;        // half select
    int kb = 2 * lh;           // VGPR0 slot: K=0|K=2 ; VGPR1: K=1|K=3

    // clamp row/col indices: garbage rows/cols are computed but never stored
    int am = tm + lr; if (am >= M) am = M - 1;
    const float* Ab = A + (long)am * lda;      // row base (for tail)
    const float* Ap = Ab + kb;

    int bnc[4];
    const float* Bp[4];
    #pragma unroll
    for (int j = 0; j < 4; ++j) {
        int bn = tn + 16 * j + lr;
        if (bn >= N) bn = N - 1;
        bnc[j] = bn;
        Bp[j] = B + (long)bn * ldbn + (long)kb * ldbk;
    }

    v8f acc0 = {}, acc1 = {}, acc2 = {}, acc3 = {};
    int K4 = K & ~3;
    #pragma unroll 2
    for (int k0 = 0; k0 < K4; k0 += 4) {
        v2f a;
        if (AM == 1) { a = *(const v2f*)(Ap + k0); }
        else         { a.x = Ap[k0]; a.y = Ap[k0 + 1]; }
        v2f b0, b1, b2, b3;
        if (BM == 2) {
            b0 = *(const v2f*)(Bp[0] + k0);
            b1 = *(const v2f*)(Bp[1] + k0);
            b2 = *(const v2f*)(Bp[2] + k0);
            b3 = *(const v2f*)(Bp[3] + k0);
        } else if (BM == 1) {
            b0.x = Bp[0][k0]; b0.y = Bp[0][k0 + 1];
            b1.x = Bp[1][k0]; b1.y = Bp[1][k0 + 1];
            b2.x = Bp[2][k0]; b2.y = Bp[2][k0 + 1];
            b3.x = Bp[3][k0]; b3.y = Bp[3][k0 + 1];
        } else {
            long o = (long)k0 * ldbk;
            b0.x = Bp[0][o]; b0.y = Bp[0][o + ldbk];
            b1.x = Bp[1][o]; b1.y = Bp[1][o + ldbk];
            b2.x = Bp[2][o]; b2.y = Bp[2][o + ldbk];
            b3.x = Bp[3][o]; b3.y = Bp[3][o + ldbk];
        }
        acc0 = __builtin_amdgcn_wmma_f32_16x16x4_f32(false, a, false, b0, (short)0, acc0, false, false);
        acc1 = __builtin_amdgcn_wmma_f32_16x16x4_f32(false, a, false, b1, (short)0, acc1, false, false);
        acc2 = __builtin_amdgcn_wmma_f32_16x16x4_f32(false, a, false, b2, (short)0, acc2, false, false);
        acc3 = __builtin_amdgcn_wmma_f32_16x16x4_f32(false, a, false, b3, (short)0, acc3, false, false);
    }
    if (K4 < K) {                              // branchless masked K-tail
        int ka0 = K4 + kb, ka1 = ka0 + 1;
        int c0 = ka0 < K ? ka0 : K - 1;
        int c1 = ka1 < K ? ka1 : K - 1;
        float m0 = ka0 < K ? 1.0f : 0.0f;
        float m1 = ka1 < K ? 1.0f : 0.0f;
        v2f a;
        a.x = m0 * Ab[c0];
        a.y = m1 * Ab[c1];
        v2f bt[4];
        #pragma unroll
        for (int j = 0; j < 4; ++j) {
            const float* Bb = B + (long)bnc[j] * ldbn;
            bt[j].x = Bb[(long)c0 * ldbk];
            bt[j].y = Bb[(long)c1 * ldbk];
        }
        acc0 = __builtin_amdgcn_wmma_f32_16x16x4_f32(false, a, false, bt[0], (short)0, acc0, false, false);
        acc1 = __builtin_amdgcn_wmma_f32_16x16x4_f32(false, a, false, bt[1], (short)0, acc1, false, false);
        acc2 = __builtin_amdgcn_wmma_f32_16x16x4_f32(false, a, false, bt[2], (short)0, acc2, false, false);
        acc3 = __builtin_amdgcn_wmma_f32_16x16x4_f32(false, a, false, bt[3], (short)0, acc3, false, false);
    }
    // D layout: VGPR r -> (M=r | M=r+8), N = lane within half
    #pragma unroll
    for (int j = 0; j < 4; ++j) {
        v8f acc = (j == 0) ? acc0 : (j == 1) ? acc1 : (j == 2) ? acc2 : acc3;
        int n = tn + 16 * j + lr;
        if (n >= N) continue;
        float bn1 = biasN  ? biasN[n]  : 0.0f;
        float bn2 = biasN2 ? biasN2[n] : 0.0f;
        #pragma unroll
        for (int r = 0; r < 8; ++r) {
            int m = tm + r + 8 * lh;
            if (m < M) {
                float v = acc[r] + bn1 + bn2;
                if (biasM) v += biasM[m];
                if (beta)  v += C[(long)m * ldc + n];
                C[(long)m * ldc + n] = v;
            }
        }
    }
}

// ================= small (language) LSTM: one workgroup, sequential =================
__global__ __launch_bounds__(1024)
void k_lstm_small(const float* __restrict__ pre,   // T x 4H (biases folded in)
                  const float* __restrict__ Whh,   // 4H x H
                  float* __restrict__ hseq) {      // T x H
    __shared__ float sh[1024];
    __shared__ float sc[1024];
    __shared__ float sg[4096];
    int tid = threadIdx.x;
    if (tid < 1024) { sh[tid] = 0.0f; sc[tid] = 0.0f; }
    __syncthreads();
    const int G4 = 4 * HID;
    for (int t = 0; t < TT; ++t) {
        for (int r = tid; r < G4; r += blockDim.x) {
            float s = pre[(long)t * G4 + r];
            const float* w = Whh + (long)r * HID;
            for (int k = 0; k < HID; k += 2) {   // HID even: paired
                v2f w2 = *(const v2f*)(w + k);
                v2f h2 = *(const v2f*)(sh + k);
                s += w2.x * h2.x + w2.y * h2.y;
            }
            sg[r] = s;
        }
        __syncthreads();
        if (tid < HID) {
            float gi = sg[tid], gf = sg[HID + tid], gg = sg[2 * HID + tid], go = sg[3 * HID + tid];
            float c = sigm(gf) * sc[tid] + sigm(gi) * tanhf(gg);
            float h = sigm(go) * tanhf(c);
            sc[tid] = c; sh[tid] = h;
            hseq[(long)t * HID + tid] = h;
        }
        __syncthreads();
    }
}

// ================= vis_xy build =================
__global__ void k_visxy(const float* __restrict__ vis, float* __restrict__ vx) {
    int idx = blockIdx.x * blockDim.x + threadIdx.x;
    if (idx >= C2 * HW) return;
    int c = idx / HW, hw = idx % HW;
    float v;
    if (c < VIS)       v = vis[idx];
    else if (c == VIS) { int w = hw % WW; v = -1.0f + 2.0f * (float)w / (float)(WW - 1); }
    else               { int h = hw / WW; v = -1.0f + 2.0f * (float)h / (float)(HH - 1); }
    vx[idx] = v;
}

// ================= attention filter =================
__global__ void k_filt(const float* __restrict__ lo, const float* __restrict__ afW,
                       const float* __restrict__ afb, float* __restrict__ filt) {
    int idx = blockIdx.x * blockDim.x + threadIdx.x;
    if (idx >= TT * C2) return;
    int t = idx / C2, r = idx % C2;
    float s = afb[r];
    const float* w = afW + (long)r * HID;
    const float* x = lo + (long)t * HID;
    for (int k = 0; k < HID; ++k) s += w[k] * x[k];
    filt[idx] = sigm(s);
}

// ================= p[t,hw] = filt[t,:] . vis_xy[:,hw] =================
__global__ void k_p(const float* __restrict__ filt, const float* __restrict__ vx,
                    float* __restrict__ p) {
    int idx = blockIdx.x * blockDim.x + threadIdx.x;
    if (idx >= TT * HW) return;
    int t = idx / HW, hw = idx % HW;
    float s = 0.0f;
    for (int c = 0; c < C2; ++c) s += filt[(long)t * C2 + c] * vx[(long)c * HW + hw];
    p[idx] = s;
}

// ================= wp[g] = Wih0[g,:] . cc_W[:,4690] =================
__global__ void k_wp(const float* __restrict__ Wih0, const float* __restrict__ ccW,
                     float* __restrict__ wp) {
    int g = blockIdx.x * blockDim.x + threadIdx.x;
    if (g >= G4MIX) return;
    float s = 0.0f;
    const float* w = Wih0 + (long)g * MIX;
    for (int m = 0; m < MIX; ++m) s += w[m] * ccW[(long)m * CCLD + (CCLD - 1)];
    wp[g] = s;
}

// ================= pre1 combine: pre1[s=hw*T+t, g] =================
__global__ void k_pre1(const float* __restrict__ Agh,  // G4 x HW
                       const float* __restrict__ Bgt,  // G4 x 16 (padded)
                       const float* __restrict__ wp,
                       const float* __restrict__ p,    // T x HW
                       const float* __restrict__ bih,
                       const float* __restrict__ bhh,
                       float* __restrict__ pre1) {     // SEQ x G4
    long idx = (long)blockIdx.x * blockDim.x + threadIdx.x;
    long total = (long)SEQ * G4MIX;
    if (idx >= total) return;
    int g = (int)(idx % G4MIX);
    long s = idx / G4MIX;
    int t = (int)(s % TT);
    int hw = (int)(s / TT);
    pre1[idx] = Agh[(long)g * HW + hw] + Bgt[g * 16 + t]
              + wp[g] * p[(long)t * HW + hw] + bih[g] + bhh[g];
}

// ================= persistent mLSTM scan =================
__device__ __forceinline__ void grid_barrier(int* bar, int target) {
    __syncthreads();
    if (threadIdx.x == 0) {
        __threadfence();
        atomicAdd(bar, 1);
        while (atomicAdd(bar, 0) < target) __builtin_amdgcn_s_sleep(2);
        __threadfence();
    }
    __syncthreads();
}

__global__ void k_scan_init(float* h, float* c, int* bar) {
    int i = blockIdx.x * blockDim.x + threadIdx.x;
    if (i < 1024) { h[i] = 0.0f; c[i] = 0.0f; }
    if (i == 0) *bar = 0;
}

// Whh (4020 x 1005) sharded across 64 persistent WGPs, each slab LDS-resident (~254KB of 320KB).
__global__ __launch_bounds__(SCAN_THREADS)
void k_mlstm_scan(const float* __restrict__ pre,   // SEQ x 4020 (biases folded)
                  const float* __restrict__ Whh,   // 4020 x 1005
                  float* __restrict__ hseq,        // SEQ x HSTR
                  float* __restrict__ g_h, float* __restrict__ g_c,
                  float* __restrict__ g_gates, int* bar) {
    const int G = SCAN_BLOCKS;
    const int rb = (G4MIX + G - 1) / G;            // 63 rows per block
    int b = blockIdx.x, tid = threadIdx.x;
    int r0 = b * rb;
    int r1 = r0 + rb; if (r1 > G4MIX) r1 = G4MIX;
    int nrows = r1 - r0;

    extern __shared__ float smem[];
    float* slab = smem;                            // nrows x RSTR weight shard (zero-padded)
    float* hl   = smem + (size_t)rb * RSTR;        // HSTR hidden copy (tail stays 0)

    for (long i = tid; i < (long)nrows * RSTR; i += blockDim.x) {
        int r = (int)(i / RSTR), c = (int)(i % RSTR);
        slab[i] = (c < HMIX) ? Whh[(long)(r0 + r) * HMIX + c] : 0.0f;
    }
    for (int j = tid; j < HSTR; j += blockDim.x) hl[j] = 0.0f;
    __syncthreads();

    int lane = tid & 31, wv = tid >> 5, nw = blockDim.x >> 5;
    int ub = (HMIX + G - 1) / G;                   // units per block in update phase
    int u0 = b * ub;
    int u1 = u0 + ub; if (u1 > HMIX) u1 = HMIX;

    int bc = 0;
    for (int s = 0; s < SEQ; ++s) {
        // gate phase: one wave per row, paired ds_load_b64 dot + wave32 xor-reduction
        for (int rr = wv; rr < nrows; rr += nw) {
            const float* w = slab + (long)rr * RSTR;
            float sum = 0.0f;
            for (int k = 2 * lane; k < RSTR; k += 64) {   // hl tail is zero
                v2f w2 = *(const v2f*)(w + k);
                v2f h2 = *(const v2f*)(hl + k);
                sum += w2.x * h2.x + w2.y * h2.y;
            }
            for (int off = 16; off; off >>= 1) sum += __shfl_xor(sum, off, 32);
            if (lane == 0)
                g_gates[r0 + rr] = sum + pre[(long)s * G4MIX + r0 + rr];
        }
        ++bc; grid_barrier(bar, bc * G);
        // update phase: each block updates its unit shard
        for (int j = u0 + tid; j < u1; j += blockDim.x) {
            float gi = g_gates[j], gf = g_gates[HMIX + j];
            float gg = g_gates[2 * HMIX + j], go = g_gates[3 * HMIX + j];
            float c = sigm(gf) * g_c[j] + sigm(gi) * tanhf(gg);
            float h = sigm(go) * tanhf(c);
            g_c[j] = c; g_h[j] = h;
            hseq[(long)s * HSTR + j] = h;
        }
        ++bc; grid_barrier(bar, bc * G);
        for (int j = tid; j < HMIX; j += blockDim.x) hl[j] = g_h[j];
        __syncthreads();
    }
}

// ================= output projection =================
__global__ void k_out(const float* __restrict__ h3, const float* __restrict__ ocW,
                      const float* __restrict__ ocb, float* __restrict__ out) {
    int p = blockIdx.x * blockDim.x + threadIdx.x;
    if (p >= HW) return;
    const float* h = h3 + (long)(SEQ - HW + p) * HSTR;
    float s = ocb[0];
    for (int c = 0; c < HMIX; ++c) s += ocW[c] * h[c];
    out[p] = s;
}

// ================= host side =================
static inline void gemm(hipStream_t st, int amode, int bmode,
                        const float* A, int lda,
                        const float* B, long ldbk, long ldbn,
                        float* C, int ldc,
                        const float* bN, const float* bN2, const float* bM,
                        int M, int N, int K, int beta) {
    int mt = (M + 15) / 16, nt = (N + 63) / 64;
    int blocks = (mt * nt + 7) / 8;                // 8 waves (256 threads) per block
#define GO(AM, BM) k_wmma_gemm_t<AM, BM><<<blocks, 256, 0, st>>>( \
        A, lda, B, ldbk, ldbn, C, ldc, bN, bN2, bM, M, N, K, beta)
    if      (amode == 1 && bmode == 2) GO(1, 2);
    else if (amode == 1 && bmode == 1) GO(1, 1);
    else if (amode == 1 && bmode == 0) GO(1, 0);
    else if (amode == 0 && bmode == 2) GO(0, 2);
    else                               GO(0, 0);
#undef GO
}

extern "C" void kernel_launch(void* const* d_in, const int* in_sizes, int n_in,
                              void* d_out, int out_size, void* d_ws, size_t ws_size,
                              hipStream_t stream) {
    const float* vis   = (const float*)d_in[0];
    const int*   lang  = (const int*)  d_in[1];
    const float* emb   = (const float*)d_in[2];
    const float* lWih  = (const float*)d_in[3];   // 2 x 4000 x 1000
    const float* lWhh  = (const float*)d_in[4];   // 2 x 4000 x 1000
    const float* lbih  = (const float*)d_in[5];   // 2 x 4000
    const float* lbhh  = (const float*)d_in[6];
    const float* mWih0 = (const float*)d_in[7];   // 4020 x 2688
    const float* mWihR = (const float*)d_in[8];   // 2 x 4020 x 1005
    const float* mWhh  = (const float*)d_in[9];   // 3 x 4020 x 1005
    const float* mbih  = (const float*)d_in[10];  // 3 x 4020
    const float* mbhh  = (const float*)d_in[11];
    const float* afW   = (const float*)d_in[12];  // 2690 x 1000
    const float* afb   = (const float*)d_in[13];
    const float* ccW   = (const float*)d_in[14];  // 2688 x 4691
    const float* ccb   = (const float*)d_in[15];
    const float* ocW   = (const float*)d_in[16];  // 1 x 1005
    const float* ocb   = (const float*)d_in[17];
    float* out = (float*)d_out;

    // ---- workspace carve (floats; all offsets even -> 8B alignment preserved) ----
    float* ws = (float*)d_ws;
    size_t off = 0;
    float* le    = ws + off; off += TT * EMB;
    float* lo1   = ws + off; off += TT * HID;
    float* lo2   = ws + off; off += TT * HID;
    float* preL  = ws + off; off += TT * 4 * HID;
    float* vx    = ws + off; off += (size_t)C2 * HW;
    float* filt  = ws + off; off += TT * C2;
    float* pbuf  = ws + off; off += TT * HW;
    float* qvis  = ws + off; off += (size_t)MIX * HW;
    float* qseq  = ws + off; off += (size_t)MIX * 16;
    float* Agh   = ws + off; off += (size_t)G4MIX * HW;
    float* Bgt   = ws + off; off += (size_t)G4MIX * 16;
    float* wp    = ws + off; off += G4MIX;
    float* preB  = ws + off; off += (size_t)SEQ * G4MIX;     // reused for layers 1..3
    float* h1seq = ws + off; off += (size_t)SEQ * HSTR;      // reused as h3seq
    float* h2seq = ws + off; off += (size_t)SEQ * HSTR;
    float* sc_h  = ws + off; off += 1024;
    float* sc_c  = ws + off; off += 1024;
    float* sc_g  = ws + off; off += 4096;
    int*   bar   = (int*)(ws + off); off += 32;
    (void)ws_size; (void)n_in; (void)in_sizes; (void)out_size;

    // ---- language path ----
    k_embed<<<(TT * EMB + 255) / 256, 256, 0, stream>>>(lang, emb, le);
    gemm(stream, 1, 2, le, EMB, lWih, 1, EMB, preL, 4 * HID,
         lbih, lbhh, nullptr, TT, 4 * HID, EMB, 0);
    k_lstm_small<<<1, 1024, 0, stream>>>(preL, lWhh, lo1);
    gemm(stream, 1, 2, lo1, HID, lWih + (size_t)4 * HID * EMB, 1, HID, preL, 4 * HID,
         lbih + 4 * HID, lbhh + 4 * HID, nullptr, TT, 4 * HID, HID, 0);
    k_lstm_small<<<1, 1024, 0, stream>>>(preL, lWhh + (size_t)4 * HID * HID, lo2);

    // ---- attention / feature algebra ----
    k_visxy<<<(C2 * HW + 255) / 256, 256, 0, stream>>>(vis, vx);
    k_filt<<<(TT * C2 + 255) / 256, 256, 0, stream>>>(lo2, afW, afb, filt);
    k_p<<<(TT * HW + 255) / 256, 256, 0, stream>>>(filt, vx, pbuf);
    // qvis[m,hw] = ccW[m,0:2690] . vx[:,hw]            (ccW rows only 4B-aligned)
    gemm(stream, 0, 0, ccW, CCLD, vx, HW, 1, qvis, HW,
         nullptr, nullptr, nullptr, MIX, HW, C2, 0);
    // qseq[m,t] = ccW[m,2690:3690].le[t] + ccW[m,3690:4690].lo2[t] + ccb[m]
    gemm(stream, 0, 2, ccW + C2, CCLD, le, 1, EMB, qseq, 16,
         nullptr, nullptr, ccb, MIX, TT, EMB, 0);
    gemm(stream, 0, 2, ccW + C2 + EMB, CCLD, lo2, 1, HID, qseq, 16,
         nullptr, nullptr, nullptr, MIX, TT, HID, 1);
    // Agh[g,hw] = Wih0[g,:] . qvis[:,hw] ; Bgt[g,t] = Wih0[g,:] . qseq[:,t]
    gemm(stream, 1, 0, mWih0, MIX, qvis, HW, 1, Agh, HW,
         nullptr, nullptr, nullptr, G4MIX, HW, MIX, 0);
    gemm(stream, 1, 0, mWih0, MIX, qseq, 16, 1, Bgt, 16,
         nullptr, nullptr, nullptr, G4MIX, TT, MIX, 0);
    k_wp<<<(G4MIX + 255) / 256, 256, 0, stream>>>(mWih0, ccW, wp);
    {
        long total = (long)SEQ * G4MIX;
        k_pre1<<<(int)((total + 255) / 256), 256, 0, stream>>>(
            Agh, Bgt, wp, pbuf, mbih, mbhh, preB);
    }

    // ---- 3-layer mLSTM scan ----
    const int rb = (G4MIX + SCAN_BLOCKS - 1) / SCAN_BLOCKS;          // 63
    const size_t shb = ((size_t)rb * RSTR + HSTR) * sizeof(float);   // ~258 KB LDS

    k_scan_init<<<4, 256, 0, stream>>>(sc_h, sc_c, bar);
    k_mlstm_scan<<<SCAN_BLOCKS, SCAN_THREADS, shb, stream>>>(
        preB, mWhh, h1seq, sc_h, sc_c, sc_g, bar);

    // pre2[s,g] = h1seq[s,:] . mWihR[0][g,:] + biases   (WihR rows 4B-aligned -> BM=1)
    gemm(stream, 1, 1, h1seq, HSTR, mWihR, 1, HMIX, preB, G4MIX,
         mbih + G4MIX, mbhh + G4MIX, nullptr, SEQ, G4MIX, HMIX, 0);
    k_scan_init<<<4, 256, 0, stream>>>(sc_h, sc_c, bar);
    k_mlstm_scan<<<SCAN_BLOCKS, SCAN_THREADS, shb, stream>>>(
        preB, mWhh + (size_t)G4MIX * HMIX, h2seq, sc_h, sc_c, sc_g, bar);

    gemm(stream, 1, 1, h2seq, HSTR, mWihR + (size_t)G4MIX * HMIX, 1, HMIX, preB, G4MIX,
         mbih + 2 * G4MIX, mbhh + 2 * G4MIX, nullptr, SEQ, G4MIX, HMIX, 0);
    k_scan_init<<<4, 256, 0, stream>>>(sc_h, sc_c, bar);
    k_mlstm_scan<<<SCAN_BLOCKS, SCAN_THREADS, shb, stream>>>(
        preB, mWhh + (size_t)2 * G4MIX * HMIX, h1seq /*h3seq*/, sc_h, sc_c, sc_g, bar);

    // ---- output ----
    k_out<<<1, 256, 0, stream>>>(h1seq, ocW, ocb, out);
}